// GNN_63745904607687
// MI455X (gfx1250) — compile-verified
//
#include <hip/hip_runtime.h>
#include <math.h>

// Problem constants (match reference)
#define NN 50000
#define EE 800000
#define DD 128   // feature dim
#define HH 256   // hidden dim
#define CC 8     // edge colours

typedef __attribute__((ext_vector_type(2))) float v2f;
typedef __attribute__((ext_vector_type(8))) float v8f;

// D = A(16x4 f32) * B(4x16 f32) + C(16x16 f32), full wave32 op.
__device__ __forceinline__ v8f wmma4(v2f a, v2f b, v8f c) {
  // 8 args: (neg_a, A, neg_b, B, c_mod, C, reuse_a, reuse_b)
  return __builtin_amdgcn_wmma_f32_16x16x4_f32(
      false, a, false, b, (short)0, c, false, false);
}

// ---------------------------------------------------------------------------
// K1: per-edge scatter-add of x[src] into agg1[colour, dst, :].
// One wave32 per edge: each lane loads a float4 of the source row (b128) and
// issues 4 global_atomic_add_f32.  x (25.6 MB) is L2-resident so the random
// gather is cheap; atomics stream to L2's atomic units.
// ---------------------------------------------------------------------------
__global__ __launch_bounds__(256) void scatter_x_kernel(
    const float* __restrict__ x, const int* __restrict__ eidx,
    const int* __restrict__ ecol, float* __restrict__ agg1) {
  const int e    = blockIdx.x * 8 + (threadIdx.x >> 5);   // 8 edges / block
  const int lane = threadIdx.x & 31;
  const int src  = eidx[e];
  const int dst  = eidx[EE + e];
  const int c    = ecol[e];
  const float4 v = *(const float4*)(x + (size_t)src * DD + lane * 4);
  float* p = agg1 + ((size_t)c * NN + dst) * DD + lane * 4;
  atomicAdd(p + 0, v.x);
  atomicAdd(p + 1, v.y);
  atomicAdd(p + 2, v.z);
  atomicAdd(p + 3, v.w);
}

// ---------------------------------------------------------------------------
// K2: fused layer-1 GEMM.  h[n,0:256] = relu(b1 + [x | agg1[0..7]] @ Bstack)
// One block = 4 waves = one 16-row M tile; wave w owns columns [64w,64w+64).
// K runs over 9 sources x 128 with V_WMMA_F32_16X16X4_F32.
// The 16x128 A tile of each source is staged once into LDS (row stride padded
// 128->132 floats so the 16 rows spread across banks) and all 4 waves read
// their A fragments with ds_load_b64 -- removes the 4x redundant global loads.
// A-frag layout (ISA 7.12.2, 32-bit A 16x4): lane(m=l&15): v0=A[m,k+2*kh],
// v1=A[m,k+2*kh+1] with kh=l>>4.  B[k,n]=W[n,k] -> contiguous b64 per lane.
// ---------------------------------------------------------------------------
__global__ __launch_bounds__(128) void layer1_gemm_kernel(
    const float* __restrict__ x, const float* __restrict__ agg1,
    const float* __restrict__ Wself1, const float* __restrict__ bself1,
    const float* __restrict__ Wconv1, float* __restrict__ h) {
  __shared__ float As[16 * 132];       // 8.45 KB, padded row stride
  const int wave = threadIdx.x >> 5;
  const int lane = threadIdx.x & 31;
  const int nlo  = lane & 15;
  const int kh   = lane >> 4;          // K half-select (0/1)
  const int m0   = blockIdx.x * 16;    // 3125 * 16 == 50000, no tail

  v8f acc[4] = {};

  #pragma unroll 1
  for (int s = 0; s < 9; ++s) {
    const float* Abase = (s == 0) ? (x + (size_t)m0 * DD)
                                  : (agg1 + ((size_t)(s - 1) * NN + m0) * DD);
    __syncthreads();                   // previous tile fully consumed
    #pragma unroll
    for (int it = 0; it < 8; ++it) {   // 1024 float4 / 128 threads
      const int j  = threadIdx.x + 128 * it;
      const int r  = j >> 5;           // 32 float4 per row
      const int c4 = j & 31;
      float4 v = *(const float4*)(Abase + (size_t)r * DD + c4 * 4);
      *(float4*)(&As[r * 132 + c4 * 4]) = v;
    }
    __syncthreads();

    const float* Bbase = (s == 0) ? Wself1
                                  : (Wconv1 + (size_t)(s - 1) * HH * DD);
    #pragma unroll 4
    for (int k = 0; k < DD; k += 4) {
      v2f a = *(const v2f*)(&As[nlo * 132 + k + 2 * kh]);   // ds_load_b64
      #pragma unroll
      for (int ct = 0; ct < 4; ++ct) {
        const int col = wave * 64 + ct * 16 + nlo;
        v2f b = *(const v2f*)(Bbase + (size_t)col * DD + k + 2 * kh);
        acc[ct] = wmma4(a, b, acc[ct]);
      }
    }
  }

  // C/D layout: VGPR j -> M = j + 8*kh, N = nlo.  Add bias, ReLU.
  #pragma unroll
  for (int ct = 0; ct < 4; ++ct) {
    const int col   = wave * 64 + ct * 16 + nlo;
    const float bia = bself1[col];
    #pragma unroll
    for (int j = 0; j < 8; ++j) {
      const int r = m0 + j + 8 * kh;
      float v = acc[ct][j] + bia;
      h[(size_t)r * HH + col] = v > 0.f ? v : 0.f;
    }
  }
}

// ---------------------------------------------------------------------------
// K3: layer-2 "multiply first".  9 waves per block (one 16-row M tile):
//   waves 0..7: t[c, rows, 0:128] = h[rows] @ W_conv2[c].T   (K=256)
//   wave 8:     out_acc[rows, 0:128] = b_self2 + h[rows] @ W_self2.T
// The 16x256 h tile is staged into LDS once (stride 256->260) and shared by
// all 9 waves (9x fewer global A loads).  Wave-uniform branch keeps EXEC
// all-ones around every WMMA.
// ---------------------------------------------------------------------------
__global__ __launch_bounds__(288) void layer2_gemm_kernel(
    const float* __restrict__ h, const float* __restrict__ Wconv2,
    const float* __restrict__ Wself2, const float* __restrict__ bself2,
    float* __restrict__ t, float* __restrict__ out_acc) {
  __shared__ float Hs[16 * 260];       // 16.64 KB, padded row stride
  const int wave = threadIdx.x / 32;   // 0..8
  const int lane = threadIdx.x & 31;
  const int nlo  = lane & 15;
  const int kh   = lane >> 4;
  const int m0   = blockIdx.x * 16;
  const bool is_self = (wave == 8);

  if (threadIdx.x < 256) {             // stage 16x256 tile: 1024 float4
    #pragma unroll
    for (int it = 0; it < 4; ++it) {
      const int j  = threadIdx.x + 256 * it;
      const int r  = j >> 6;           // 64 float4 per row
      const int c4 = j & 63;
      float4 v = *(const float4*)(h + (size_t)(m0 + r) * HH + c4 * 4);
      *(float4*)(&Hs[r * 260 + c4 * 4]) = v;
    }
  }
  __syncthreads();

  const float* Bbase = is_self ? Wself2 : (Wconv2 + (size_t)wave * DD * HH);

  v8f acc[8] = {};

  #pragma unroll 2
  for (int k = 0; k < HH; k += 4) {
    v2f a = *(const v2f*)(&Hs[nlo * 260 + k + 2 * kh]);     // ds_load_b64
    #pragma unroll
    for (int ct = 0; ct < 8; ++ct) {
      const int col = ct * 16 + nlo;
      v2f b = *(const v2f*)(Bbase + (size_t)col * HH + k + 2 * kh);
      acc[ct] = wmma4(a, b, acc[ct]);
    }
  }

  if (is_self) {
    #pragma unroll
    for (int ct = 0; ct < 8; ++ct) {
      const int col   = ct * 16 + nlo;
      const float bia = bself2[col];
      #pragma unroll
      for (int j = 0; j < 8; ++j)
        out_acc[(size_t)(m0 + j + 8 * kh) * DD + col] = acc[ct][j] + bia;
    }
  } else {
    float* tc = t + (size_t)wave * NN * DD;
    #pragma unroll
    for (int ct = 0; ct < 8; ++ct) {
      const int col = ct * 16 + nlo;
      #pragma unroll
      for (int j = 0; j < 8; ++j)
        tc[(size_t)(m0 + j + 8 * kh) * DD + col] = acc[ct][j];
    }
  }
}

// ---------------------------------------------------------------------------
// K4: per-edge scatter-add of pre-transformed message t[colour, src] into
// out_acc[dst].  One wave32 per edge, float4 per lane (128 floats/edge).
// ---------------------------------------------------------------------------
__global__ __launch_bounds__(256) void scatter_t_kernel(
    const float* __restrict__ t, const int* __restrict__ eidx,
    const int* __restrict__ ecol, float* __restrict__ out_acc) {
  const int e    = blockIdx.x * 8 + (threadIdx.x >> 5);
  const int lane = threadIdx.x & 31;
  const int src  = eidx[e];
  const int dst  = eidx[EE + e];
  const int c    = ecol[e];
  const float4 v = *(const float4*)(t + ((size_t)c * NN + src) * DD + lane * 4);
  float* p = out_acc + (size_t)dst * DD + lane * 4;
  atomicAdd(p + 0, v.x);
  atomicAdd(p + 1, v.y);
  atomicAdd(p + 2, v.z);
  atomicAdd(p + 3, v.w);
}

// K5: out = sigmoid(acc - 10), vectorized float4.
__global__ __launch_bounds__(256) void finalize_kernel(
    const float4* __restrict__ acc, float4* __restrict__ out, int n4) {
  const int i = blockIdx.x * 256 + threadIdx.x;
  if (i < n4) {
    float4 a = acc[i];
    float4 r;
    r.x = 1.f / (1.f + __expf(10.f - a.x));
    r.y = 1.f / (1.f + __expf(10.f - a.y));
    r.z = 1.f / (1.f + __expf(10.f - a.z));
    r.w = 1.f / (1.f + __expf(10.f - a.w));
    out[i] = r;
  }
}

// ---------------------------------------------------------------------------
// kernel_launch
// Inputs (setup_inputs order):
//  0 x[N,D] f32, 1 edge_index[2,E] int, 2 edge_colour[E] int,
//  3 W_self1[H,D], 4 b_self1[H], 5 W_self2[D,H], 6 b_self2[D],
//  7 W_conv1[C,H,D], 8 W_conv2[C,D,H]
// Workspace layout (fp32): agg1 [C,N,D] 204.8MB | h [N,H] 51.2MB |
// out_acc [N,D] 25.6MB.  t [C,N,D] reuses agg1's region (agg1 is dead after
// layer1 GEMM).  Total required ws: 281.6 MB.
// ---------------------------------------------------------------------------
extern "C" void kernel_launch(void* const* d_in, const int* in_sizes, int n_in,
                              void* d_out, int out_size, void* d_ws,
                              size_t ws_size, hipStream_t stream) {
  const float* x      = (const float*)d_in[0];
  const int*   eidx   = (const int*)d_in[1];
  const int*   ecol   = (const int*)d_in[2];
  const float* Wself1 = (const float*)d_in[3];
  const float* bself1 = (const float*)d_in[4];
  const float* Wself2 = (const float*)d_in[5];
  const float* bself2 = (const float*)d_in[6];
  const float* Wconv1 = (const float*)d_in[7];
  const float* Wconv2 = (const float*)d_in[8];
  float* out = (float*)d_out;

  float* agg1 = (float*)d_ws;                       // C*N*D floats
  float* h    = agg1 + (size_t)CC * NN * DD;        // N*H floats
  float* oacc = h + (size_t)NN * HH;                // N*D floats
  float* t    = agg1;                               // reuse agg1 region

  hipMemsetAsync(agg1, 0, (size_t)CC * NN * DD * sizeof(float), stream);

  scatter_x_kernel<<<EE / 8, 256, 0, stream>>>(x, eidx, ecol, agg1);
  layer1_gemm_kernel<<<NN / 16, 128, 0, stream>>>(x, agg1, Wself1, bself1,
                                                  Wconv1, h);
  layer2_gemm_kernel<<<NN / 16, 288, 0, stream>>>(h, Wconv2, Wself2, bself2,
                                                  t, oacc);
  scatter_t_kernel<<<EE / 8, 256, 0, stream>>>(t, eidx, ecol, oacc);
  finalize_kernel<<<(NN * DD / 4 + 255) / 256, 256, 0, stream>>>(
      (const float4*)oacc, (float4*)out, NN * DD / 4);
}